// FactorizationMachineModel_withGCN_82824149336601
// MI455X (gfx1250) — compile-verified
//
#include <hip/hip_runtime.h>
#include <hip/hip_bf16.h>

typedef __attribute__((ext_vector_type(2))) float v2f;
typedef __attribute__((ext_vector_type(8))) float v8f;

#define EMBED_DIM 64

// ---------------------------------------------------------------------------
// 1) init: deg[i] = 1.0 (self loop), agg = 0
// ---------------------------------------------------------------------------
__global__ void init_kernel(float* __restrict__ deg, float* __restrict__ agg,
                            int n, int total) {
    int i = blockIdx.x * blockDim.x + threadIdx.x;
    if (i < total) agg[i] = 0.0f;
    if (i < n)     deg[i] = 1.0f;
}

// ---------------------------------------------------------------------------
// 2) degree: one thread per edge, atomic add on dst
// ---------------------------------------------------------------------------
__global__ void degree_kernel(const long long* __restrict__ ei,
                              float* __restrict__ deg, int E) {
    int e = blockIdx.x * blockDim.x + threadIdx.x;
    if (e < E) {
        int d = (int)ei[(size_t)E + e];
        atomicAdd(&deg[d], 1.0f);
    }
}

// ---------------------------------------------------------------------------
// 3) dinv = rsqrt(deg), in place
// ---------------------------------------------------------------------------
__global__ void rsqrt_kernel(float* __restrict__ deg, int n) {
    int i = blockIdx.x * blockDim.x + threadIdx.x;
    if (i < n) deg[i] = rsqrtf(deg[i]);
}

// ---------------------------------------------------------------------------
// 4) edge scatter: one wave32 per edge; lane handles dims [2*lane, 2*lane+1].
//    agg[dst] += gcn_weight[src] * dinv[src]*dinv[dst]   (L2-resident atomics)
// ---------------------------------------------------------------------------
__global__ __launch_bounds__(256)
void scatter_kernel(const long long* __restrict__ ei,
                    const float* __restrict__ w,
                    const float* __restrict__ dinv,
                    float* __restrict__ agg, int E) {
    int gtid = blockIdx.x * blockDim.x + threadIdx.x;
    int edge = gtid >> 5;
    int lane = gtid & 31;
    if (edge >= E) return;
    int s = (int)ei[edge];
    int d = (int)ei[(size_t)E + edge];
    float norm = dinv[s] * dinv[d];
    const float2* wrow = (const float2*)(w + (size_t)s * EMBED_DIM);
    float2 v = wrow[lane];
    float* arow = agg + (size_t)d * EMBED_DIM + lane * 2;
    atomicAdd(arow,     v.x * norm);
    atomicAdd(arow + 1, v.y * norm);
}

// ---------------------------------------------------------------------------
// 5) Fused FM head with WMMA.
//    Block = 128 threads (4 waves), 64 pairs/block (16 pairs per wave).
//    Phase 1: stage emb[node] = agg + w*dinv^2 + bias for 64 pairs x 2 nodes
//             into LDS (32 KB).
//    Phase 2: per wave, D = E0 (16x4 f32) x E1^T (4x16 f32) accumulated over
//             K=64 via 16x V_WMMA_F32_16X16X4_F32; dot(e0_m, e1_m) = D[m][m].
//    A/B lane layout (16x4 f32, CDNA5): lane L holds row M=L%16;
//      VGPR0 = K{0 | 2}, VGPR1 = K{1 | 3} for lane halves {0-15 | 16-31}.
//    C/D 16x16 f32 layout: VGPR r, lanes 0-15 -> M=r, lanes 16-31 -> M=r+8.
//      diag(m<8)  = lane m,    VGPR m
//      diag(m>=8) = lane m+16, VGPR m-8   (extracted via LDS round-trip)
// ---------------------------------------------------------------------------
#define PAIRS_PER_BLOCK 64
#define FM_THREADS 128

__global__ __launch_bounds__(FM_THREADS)
void fm_kernel(const float* __restrict__ w,
               const float* __restrict__ bias,
               const float* __restrict__ lw,
               const float* __restrict__ lbias,
               const float* __restrict__ dinv,
               const float* __restrict__ agg,
               const long long* __restrict__ pairs,
               float* __restrict__ out, int B) {
    __shared__ float emb[PAIRS_PER_BLOCK * 2 * EMBED_DIM]; // 32 KB
    __shared__ float diag[4 * 256];                        // 4 KB

    int blockStart = blockIdx.x * PAIRS_PER_BLOCK;
    int tid = threadIdx.x;

    // Phase 1: build node embeddings for this block's pairs into LDS.
    for (int j = tid; j < PAIRS_PER_BLOCK * 2 * EMBED_DIM; j += FM_THREADS) {
        int d     = j & (EMBED_DIM - 1);
        int sw    = j >> 6;          // slot*2 + which
        int slot  = sw >> 1;
        int which = sw & 1;
        int node  = (int)pairs[(size_t)(blockStart + slot) * 2 + which];
        float di  = dinv[node];
        emb[j] = agg[(size_t)node * EMBED_DIM + d]
               + w[(size_t)node * EMBED_DIM + d] * di * di
               + bias[d];
    }
    __syncthreads();

    // Phase 2: WMMA over K=64 in chunks of 4.
    int waveId = tid >> 5;
    int lane   = tid & 31;
    int m      = lane & 15;
    int koff   = (lane < 16) ? 0 : 2;
    int slot0  = waveId * 16;

    const float* e0 = &emb[((slot0 + m) * 2 + 0) * EMBED_DIM];
    const float* e1 = &emb[((slot0 + m) * 2 + 1) * EMBED_DIM];

    v8f c = {};
#pragma unroll
    for (int k0 = 0; k0 < EMBED_DIM; k0 += 4) {
        v2f a = { e0[k0 + koff], e0[k0 + koff + 1] };
        v2f b = { e1[k0 + koff], e1[k0 + koff + 1] };
        // 8 args: (neg_a, A, neg_b, B, c_mod, C, reuse_a, reuse_b)
        c = __builtin_amdgcn_wmma_f32_16x16x4_f32(
                false, a, false, b, (short)0, c, false, false);
    }

    // Dump accumulator to LDS and pull the diagonal.
    float* dl = &diag[waveId * 256];
#pragma unroll
    for (int r = 0; r < 8; ++r) dl[r * 32 + lane] = c[r];
    __syncthreads();

    if (lane < 16) {
        int p = blockStart + slot0 + lane;
        if (p < B) {
            float dv = (lane < 8) ? dl[lane * 32 + lane]
                                  : dl[(lane - 8) * 32 + (lane + 16)];
            int p0 = (int)pairs[(size_t)p * 2];
            int p1 = (int)pairs[(size_t)p * 2 + 1];
            out[p] = dv + lw[p0] + lw[p1] + lbias[0];
        }
    }
}

// ---------------------------------------------------------------------------
// Launcher
// ---------------------------------------------------------------------------
extern "C" void kernel_launch(void* const* d_in, const int* in_sizes, int n_in,
                              void* d_out, int out_size, void* d_ws, size_t ws_size,
                              hipStream_t stream) {
    const float*     gcn_w  = (const float*)d_in[0];     // [N, 64]
    const float*     gcn_b  = (const float*)d_in[1];     // [64]
    const float*     lin_w  = (const float*)d_in[2];     // [N, 1]
    const float*     lin_b  = (const float*)d_in[3];     // [1]
    const long long* ei     = (const long long*)d_in[4]; // [2, E] int64
    const long long* pairs  = (const long long*)d_in[5]; // [B, 2] int64
    float*           out    = (float*)d_out;             // [B]

    int N = in_sizes[0] / EMBED_DIM;
    int E = in_sizes[4] / 2;
    int B = in_sizes[5] / 2;

    float* deg = (float*)d_ws;           // N floats (becomes dinv in place)
    float* agg = deg + N;                // N*64 floats

    int total = N * EMBED_DIM;

    // 1) init
    {
        int threads = 256;
        int blocks = (total + threads - 1) / threads;
        init_kernel<<<blocks, threads, 0, stream>>>(deg, agg, N, total);
    }
    // 2) degree
    {
        int threads = 256;
        int blocks = (E + threads - 1) / threads;
        degree_kernel<<<blocks, threads, 0, stream>>>(ei, deg, E);
    }
    // 3) rsqrt
    {
        int threads = 256;
        int blocks = (N + threads - 1) / threads;
        rsqrt_kernel<<<blocks, threads, 0, stream>>>(deg, N);
    }
    // 4) edge scatter (wave per edge)
    {
        long long tthreads = (long long)E * 32;
        int threads = 256;
        int blocks = (int)((tthreads + threads - 1) / threads);
        scatter_kernel<<<blocks, threads, 0, stream>>>(ei, gcn_w, deg, agg, E);
    }
    // 5) fused FM head (WMMA)
    {
        int blocks = (B + PAIRS_PER_BLOCK - 1) / PAIRS_PER_BLOCK;
        fm_kernel<<<blocks, FM_THREADS, 0, stream>>>(
            gcn_w, gcn_b, lin_w, lin_b, deg, agg, pairs, out, B);
    }
}